// BasisLinear_71846212927837
// MI455X (gfx1250) — compile-verified
//
#include <hip/hip_runtime.h>

typedef float v2f __attribute__((ext_vector_type(2)));
typedef float v8f __attribute__((ext_vector_type(8)));

// ---------------------------------------------------------------------------
// Kernel 1: materialize Wt (B-operand of the WMMA GEMM) in the swizzled
// fragment layout, plus the fused bias vector.
//   W[n,k] = sum_r basis[r,0,p,0,q] * coeffs[r,O,0,I,0],  n=O*4+p, k=I*4+q
// Swizzle: element (k,n) stored at ws[(k>>1)*512 + 2*n + (k&1)], so that a
// lane's 8-byte LDS read returns {W[.,k], W[.,k+1]} for its column n.
// bias[n] stored at ws[65536 + n].
// ---------------------------------------------------------------------------
__global__ __launch_bounds__(256) void build_w_kernel(
    const float* __restrict__ basis,    // (16,1,4,1,4)  -> r*16 + p*4 + q
    const float* __restrict__ coeffs,   // (16,64,1,64,1)-> r*4096 + O*64 + I
    const float* __restrict__ bbias,    // (4,1,4)       -> r*4 + p
    const float* __restrict__ cbias,    // (4,64,1)      -> r*64 + O
    float* __restrict__ ws)
{
    const int idx = blockIdx.x * 256 + threadIdx.x;   // 0..65535
    const int n = idx >> 8;                           // 0..255
    const int k = idx & 255;                          // 0..255
    const int O = n >> 2, p = n & 3;
    const int I = k >> 2, q = k & 3;

    float acc = 0.0f;
#pragma unroll
    for (int r = 0; r < 16; ++r)
        acc += basis[r * 16 + p * 4 + q] * coeffs[r * 4096 + O * 64 + I];

    ws[(k >> 1) * 512 + 2 * n + (k & 1)] = acc;

    if (k == 0) {
        float b = 0.0f;
#pragma unroll
        for (int r = 0; r < 4; ++r)
            b += bbias[r * 4 + p] * cbias[r * 64 + O];
        ws[65536 + n] = b;
    }
}

// ---------------------------------------------------------------------------
// Kernel 2: out(65536x256) = X(65536x256) * W^T + bias, via
// v_wmma_f32_16x16x4_f32.  512 threads = 16 wave32 per workgroup.
// Whole W^T (256 KB) staged in LDS once per workgroup; each wave owns one
// 16-row b-tile and all 16 n-tiles (16 x v8f accumulators), so X streams
// from HBM exactly once.
// ---------------------------------------------------------------------------
__global__ __launch_bounds__(512, 1) void basis_linear_gemm_kernel(
    const float* __restrict__ X,     // (65536, 256)
    const float* __restrict__ wsW,   // swizzled Wt + bias (see kernel 1)
    float* __restrict__ out)         // (65536, 256)
{
    __shared__ __align__(16) float ldsB[65536];       // 256 KB of 320 KB WGP LDS

    // Cooperative fill: 16384 float4 / 512 threads = 32 per thread, coalesced.
    {
        const float4* src = (const float4*)wsW;
        float4*       dst = (float4*)ldsB;
        for (int i = threadIdx.x; i < 16384; i += 512)
            dst[i] = src[i];
    }
    __syncthreads();

    const int wave = threadIdx.x >> 5;                // 0..15
    const int lane = threadIdx.x & 31;
    const int nn   = lane & 15;                       // column / row-in-tile
    const int hi   = lane >> 4;                       // half-wave selector

    const int b0 = blockIdx.x * 256 + wave * 16;      // this wave's b-tile

    // A-fragment source: lane holds row (b0+nn), K pair (k0 + 2*hi, +1)
    const float* xrow = X + (b0 + nn) * 256 + hi * 2;

    // Per-lane bias values (depend only on n = nt*16+nn)
    float bvals[16];
#pragma unroll
    for (int nt = 0; nt < 16; ++nt)
        bvals[nt] = wsW[65536 + nt * 16 + nn];

    v8f acc[16];
#pragma unroll
    for (int nt = 0; nt < 16; ++nt)
        acc[nt] = (v8f)(0.0f);

    const v2f* B2 = (const v2f*)ldsB;
    const int  bsel = hi * 256 + nn;                  // invariant part of B index

    for (int c = 0; c < 64; ++c) {                    // K chunks of 4
        v2f va = *(const v2f*)(xrow + c * 4);         // global, 8B aligned
        const v2f* brow = B2 + (c * 512 + bsel);      // (c*2+hi)*256 + nn
#pragma unroll
        for (int nt = 0; nt < 16; ++nt) {
            v2f vb = brow[nt * 16];                   // ds_load_b64
            acc[nt] = __builtin_amdgcn_wmma_f32_16x16x4_f32(
                /*neg_a=*/false, va, /*neg_b=*/false, vb,
                /*c_mod=*/(short)0, acc[nt],
                /*reuse_a=*/false, /*reuse_b=*/false);
        }
    }

    // D layout: VGPR v -> rows M=v (lanes 0-15) / M=v+8 (lanes 16-31), N=nn.
#pragma unroll
    for (int nt = 0; nt < 16; ++nt) {
        const int ncol = nt * 16 + nn;
#pragma unroll
        for (int v = 0; v < 8; ++v) {
            const int brow_out = b0 + v + hi * 8;
            out[brow_out * 256 + ncol] = acc[nt][v] + bvals[nt];
        }
    }
}

extern "C" void kernel_launch(void* const* d_in, const int* in_sizes, int n_in,
                              void* d_out, int out_size, void* d_ws, size_t ws_size,
                              hipStream_t stream) {
    const float* x      = (const float*)d_in[0];
    const float* basis  = (const float*)d_in[1];
    const float* coeffs = (const float*)d_in[2];
    const float* bbias  = (const float*)d_in[3];
    const float* cbias  = (const float*)d_in[4];
    float* out = (float*)d_out;
    float* ws  = (float*)d_ws;   // needs (65536 + 256) * 4 = 257 KB

    build_w_kernel<<<256, 256, 0, stream>>>(basis, coeffs, bbias, cbias, ws);
    basis_linear_gemm_kernel<<<256, 512, 0, stream>>>(x, ws, out);
}